// WideAndDeep_16673063043692
// MI455X (gfx1250) — compile-verified
//
#include <hip/hip_runtime.h>

// ---------------------------------------------------------------------------
// Wide&Deep forward, MI455X (gfx1250, wave32, WMMA fp32 path).
//   B=16384 rows, D=128, F1=64, F2=32, N=100000.
// Memory-bound on the 8.4MB embedding gather -> keep everything fp32 and use
// V_WMMA_F32_16X16X4_F32 for the two dense layers (precision-first; matrix
// throughput is not the bottleneck at this arithmetic intensity).
// ---------------------------------------------------------------------------

typedef __attribute__((ext_vector_type(2))) float v2f;
typedef __attribute__((ext_vector_type(8))) float v8f;

#define B_SZ 16384
#define D_SZ 128
#define F1_SZ 64
#define F2_SZ 32
#define WAVES_PER_BLOCK 4

__global__ __launch_bounds__(32 * WAVES_PER_BLOCK)
void wide_deep_fwd(const int* __restrict__ x,
                   const float* __restrict__ emb,
                   const float* __restrict__ W1,
                   const float* __restrict__ b1,
                   const float* __restrict__ W2,
                   const float* __restrict__ b2,
                   const float* __restrict__ Wo,
                   const float* __restrict__ bo,
                   float* __restrict__ out)
{
    // per-wave h1 tile: 16 rows x F1 cols, row-major
    __shared__ float h1lds[WAVES_PER_BLOCK][16 * F1_SZ];

    const int lane    = threadIdx.x & 31;
    const int wave    = threadIdx.x >> 5;
    const int tile    = blockIdx.x * WAVES_PER_BLOCK + wave; // 16-row batch tile
    const int r0      = tile * 16;
    const int halfSel = lane >> 4;   // 0: lanes 0-15, 1: lanes 16-31
    const int mrow    = lane & 15;   // row (A) or column (B / C) id within tile

    // embedding row index for this lane's A-matrix row
    const int xi = x[r0 + mrow];

    // ---- Layer 1: h1[16x64] = emb[x[rows]] (16x128) @ W1^T --------------------
    // A chunk c (16x4 fp32 layout): lanes 0-15 -> M=lane,   K = 4c+{0,1}
    //                               lanes16-31 -> M=lane-16, K = 4c+{2,3}
    // => one b64 load per lane at emb[xi*D + 4c + 2*halfSel].
    // B chunk: B[k][n] = W1[n][k]; same half-split of K across the lane halves.
    const float* aBase  = emb + (size_t)xi * D_SZ + 2 * halfSel;
    const float* w1Base = W1 + (size_t)mrow * D_SZ + 2 * halfSel;

    v8f acc1[4] = {};   // four 16-wide N-tiles covering F1=64

#pragma unroll
    for (int c = 0; c < D_SZ / 4; ++c) {
        v2f a = *(const v2f*)(aBase + 4 * c);
#pragma unroll
        for (int t = 0; t < 4; ++t) {
            v2f b = *(const v2f*)(w1Base + t * 16 * D_SZ + 4 * c);
            acc1[t] = __builtin_amdgcn_wmma_f32_16x16x4_f32(
                /*neg_a=*/false, a, /*neg_b=*/false, b,
                /*c_mod=*/(short)0, acc1[t],
                /*reuse_a=*/false, /*reuse_b=*/false);
        }
    }

    // add b1, spill h1 tile to LDS row-major [16][64] (C layout: VGPR j holds
    // row M=j+8*halfSel, column n=t*16+mrow)
    float* hs = h1lds[wave];
#pragma unroll
    for (int t = 0; t < 4; ++t) {
        const float bias = b1[t * 16 + mrow];
#pragma unroll
        for (int j = 0; j < 8; ++j) {
            const int M = j + 8 * halfSel;
            hs[M * F1_SZ + t * 16 + mrow] = acc1[t][j] + bias;
        }
    }
    __syncthreads();

    // ---- Layer 2: h2[16x32] = h1 (16x64) @ W2^T ------------------------------
    const float* a2Base = hs + mrow * F1_SZ + 2 * halfSel;          // LDS, A layout
    const float* w2Base = W2 + (size_t)mrow * F1_SZ + 2 * halfSel;  // B layout

    v8f acc2[2] = {};   // two 16-wide N-tiles covering F2=32

#pragma unroll
    for (int c = 0; c < F1_SZ / 4; ++c) {
        v2f a = *(const v2f*)(a2Base + 4 * c);   // ds_load_b64
#pragma unroll
        for (int t = 0; t < 2; ++t) {
            v2f b = *(const v2f*)(w2Base + t * 16 * F1_SZ + 4 * c);
            acc2[t] = __builtin_amdgcn_wmma_f32_16x16x4_f32(
                false, a, false, b, (short)0, acc2[t], false, false);
        }
    }

    // ---- Output layer: y[M] = sum_n (h2[M][n]+b2[n]) * Wo[n] + Wo[32+x[M]] + bo
    const float wo0  = Wo[mrow];             // n = mrow        (tile 0)
    const float wo1  = Wo[16 + mrow];        // n = 16 + mrow   (tile 1)
    const float b20  = b2[mrow];
    const float b21  = b2[16 + mrow];
    const float bout = bo[0];

    float p[8];
#pragma unroll
    for (int j = 0; j < 8; ++j)
        p[j] = (acc2[0][j] + b20) * wo0 + (acc2[1][j] + b21) * wo1;

    // reduce over the 16 lanes of each half-wave (xor masks <16 never cross halves)
#pragma unroll
    for (int j = 0; j < 8; ++j) {
        p[j] += __shfl_xor(p[j], 8, 32);
        p[j] += __shfl_xor(p[j], 4, 32);
        p[j] += __shfl_xor(p[j], 2, 32);
        p[j] += __shfl_xor(p[j], 1, 32);
    }

    // lanes 0 / 16 own rows r0..r0+7 / r0+8..r0+15
    if (mrow == 0) {
        const int base = r0 + 8 * halfSel;
#pragma unroll
        for (int j = 0; j < 8; ++j) {
            const int row = base + j;
            out[row] = p[j] + Wo[F2_SZ + x[row]] + bout;
        }
    }
}

extern "C" void kernel_launch(void* const* d_in, const int* in_sizes, int n_in,
                              void* d_out, int out_size, void* d_ws, size_t ws_size,
                              hipStream_t stream) {
    (void)in_sizes; (void)n_in; (void)out_size; (void)d_ws; (void)ws_size;
    const int*   x   = (const int*)  d_in[0];
    const float* emb = (const float*)d_in[1];
    const float* W1  = (const float*)d_in[2];
    const float* b1  = (const float*)d_in[3];
    const float* W2  = (const float*)d_in[4];
    const float* b2  = (const float*)d_in[5];
    const float* Wo  = (const float*)d_in[6];
    const float* bo  = (const float*)d_in[7];
    float*       out = (float*)d_out;

    const int tiles  = B_SZ / 16;                 // 1024 wave-tiles
    const int blocks = tiles / WAVES_PER_BLOCK;   // 256 workgroups
    wide_deep_fwd<<<blocks, 32 * WAVES_PER_BLOCK, 0, stream>>>(
        x, emb, W1, b1, W2, b2, Wo, bo, out);
}